// HeteroGNNModel_57011395887283
// MI455X (gfx1250) — compile-verified
//
#include <hip/hip_runtime.h>

typedef __attribute__((ext_vector_type(2))) float v2f;
typedef __attribute__((ext_vector_type(8))) float v8f;

#define DFEAT 128
#define LDS_PITCH 132   // 128 + 4 pad -> consecutive A rows land on different LDS banks

// ---------------------------------------------------------------------------
// Edge scatter: msg[dst] += x[src], cnt[dst] += 1  (mean-agg numerator/denom)
// One wave32 per edge; lane l moves features 4l..4l+3 as a float4.
// ---------------------------------------------------------------------------
__global__ __launch_bounds__(256)
void scatter_add_kernel(const float* __restrict__ x,
                        const int* __restrict__ src,
                        const int* __restrict__ dst,
                        float* __restrict__ msg,
                        float* __restrict__ cnt,
                        int nedges) {
    int warp = (int)((blockIdx.x * blockDim.x + threadIdx.x) >> 5);
    int lane = threadIdx.x & 31;
    if (warp >= nedges) return;
    int s = src[warp];
    int d = dst[warp];
    const float4 v = ((const float4*)(x + (size_t)s * DFEAT))[lane];
    float* mrow = msg + (size_t)d * DFEAT + lane * 4;
    atomicAdd(mrow + 0, v.x);
    atomicAdd(mrow + 1, v.y);
    atomicAdd(mrow + 2, v.z);
    atomicAdd(mrow + 3, v.w);
    if (lane == 0) atomicAdd(cnt + d, 1.0f);
}

// ---------------------------------------------------------------------------
// Combine the two "root" linear weights that act on the same x_dst:
//   Wc = Wa + Wb (128x128), bc = ba + bb (128)
// ---------------------------------------------------------------------------
__global__ __launch_bounds__(256)
void combine_w_kernel(const float* __restrict__ Wa, const float* __restrict__ Wb,
                      const float* __restrict__ ba, const float* __restrict__ bb,
                      float* __restrict__ Wc, float* __restrict__ bc) {
    int i = blockIdx.x * blockDim.x + threadIdx.x;
    if (i < DFEAT * DFEAT) Wc[i] = Wa[i] + Wb[i];
    if (i < DFEAT)         bc[i] = ba[i] + bb[i];
}

// ---------------------------------------------------------------------------
// Fused SAGE update for one node type:
//   out = leaky_relu( sum_s  A_s @ W_s^T  + bias )
// A0 (with cnt0) and optional A1 (with cnt1) are mean-normalized aggregates;
// A2 is the raw x_dst (no normalization). W are [128,128] row-major, so
// B[k][n] = W[n][k] : lane n reads row n of W -- contiguous float2 per step.
// Block = 256 thr = 8 waves; blockIdx.x = 16-row M tile; wave w = N tile w.
// Uses V_WMMA_F32_16X16X4_F32 (f32 matrix pipe, full precision).
// ---------------------------------------------------------------------------
__global__ __launch_bounds__(256)
void sage_gemm_kernel(const float* __restrict__ A0, const float* __restrict__ cnt0,
                      const float* __restrict__ W0,
                      const float* __restrict__ A1, const float* __restrict__ cnt1,
                      const float* __restrict__ W1,
                      const float* __restrict__ A2, const float* __restrict__ W2,
                      const float* __restrict__ bias,
                      float* __restrict__ out, int M) {
    __shared__ float smem[3 * 16 * LDS_PITCH];
    const int mbase = blockIdx.x * 16;
    const int tid = threadIdx.x;

    // Stage (normalized) 16x128 A tiles for each active source into LDS.
    for (int s = 0; s < 3; ++s) {
        const float* A = (s == 0) ? A0 : (s == 1) ? A1 : A2;
        if (A == nullptr) continue;                       // uniform branch
        const float* C = (s == 0) ? cnt0 : (s == 1) ? cnt1 : nullptr;
        float* stg = smem + s * 16 * LDS_PITCH;
        for (int e = tid; e < 16 * DFEAT; e += 256) {
            int r = e >> 7, c = e & 127;
            int m = mbase + r;
            float v = 0.0f;
            if (m < M) {
                v = A[(size_t)m * DFEAT + c];
                if (C) v *= 1.0f / fmaxf(C[m], 1.0f);     // mean = sum / max(cnt,1)
            }
            stg[r * LDS_PITCH + c] = v;
        }
    }
    __syncthreads();

    const int wave = tid >> 5;      // N tile (0..7)
    const int lane = tid & 31;
    const int half = lane >> 4;     // selects K pair per ISA A/B layout
    const int l16  = lane & 15;
    const int nbase = wave * 16;

    v8f acc = {0.f, 0.f, 0.f, 0.f, 0.f, 0.f, 0.f, 0.f};

    for (int s = 0; s < 3; ++s) {
        const float* A = (s == 0) ? A0 : (s == 1) ? A1 : A2;
        if (A == nullptr) continue;
        const float* W = (s == 0) ? W0 : (s == 1) ? W1 : W2;
        const float* arow = smem + s * 16 * LDS_PITCH + l16 * LDS_PITCH; // A row = lane M
        const float* brow = W + (size_t)(nbase + l16) * DFEAT;           // B col = lane N
        #pragma unroll
        for (int k = 0; k < DFEAT; k += 4) {
            v2f a = *(const v2f*)(arow + k + 2 * half);   // ds_read_b64
            v2f b = *(const v2f*)(brow + k + 2 * half);   // global b64
            acc = __builtin_amdgcn_wmma_f32_16x16x4_f32(
                false, a, false, b, (short)0, acc, false, false);
        }
    }

    // C layout: VGPR r -> (M = r + 8*half, N = l16)
    #pragma unroll
    for (int r = 0; r < 8; ++r) {
        int m = mbase + r + 8 * half;
        if (m < M) {
            float v = acc[r] + bias[nbase + l16];
            v = (v > 0.0f) ? v : 0.01f * v;               // leaky_relu
            out[(size_t)m * DFEAT + nbase + l16] = v;
        }
    }
}

// ---------------------------------------------------------------------------
// Final projection: logits[M,16] = X[M,128] @ Wlin[16,128]^T + blin
// One wave per 16-row tile; single N tile of 16.
// ---------------------------------------------------------------------------
__global__ __launch_bounds__(32)
void final_linear_kernel(const float* __restrict__ X, const float* __restrict__ W,
                         const float* __restrict__ b, float* __restrict__ out, int M) {
    const int mbase = blockIdx.x * 16;
    const int lane = threadIdx.x & 31;
    const int half = lane >> 4;
    const int l16  = lane & 15;

    const float* arow = X + (size_t)(mbase + l16) * DFEAT;   // M exact multiple of 16
    const float* brow = W + (size_t)l16 * DFEAT;

    v8f acc = {0.f, 0.f, 0.f, 0.f, 0.f, 0.f, 0.f, 0.f};
    #pragma unroll
    for (int k = 0; k < DFEAT; k += 4) {
        v2f a = *(const v2f*)(arow + k + 2 * half);
        v2f bb = *(const v2f*)(brow + k + 2 * half);
        acc = __builtin_amdgcn_wmma_f32_16x16x4_f32(
            false, a, false, bb, (short)0, acc, false, false);
    }
    #pragma unroll
    for (int r = 0; r < 8; ++r) {
        int m = mbase + r + 8 * half;
        if (m < M) out[(size_t)m * 16 + l16] = acc[r] + b[l16];
    }
}

// ---------------------------------------------------------------------------
extern "C" void kernel_launch(void* const* d_in, const int* in_sizes, int n_in,
                              void* d_out, int out_size, void* d_ws, size_t ws_size,
                              hipStream_t stream) {
    const int NP = 100000, NA = 50000;
    const int EPP = 600000, EWR = 400000, ERW = 400000;

    const float* x_paper  = (const float*)d_in[0];
    const float* x_author = (const float*)d_in[1];
    const int* pp_s = (const int*)d_in[2];
    const int* pp_d = (const int*)d_in[3];
    const int* wr_s = (const int*)d_in[4];
    const int* wr_d = (const int*)d_in[5];
    const int* rw_s = (const int*)d_in[6];
    const int* rw_d = (const int*)d_in[7];
    // params flattened in pytree (sorted-key) order:
    // Wlin, blin, then layers[0..1] x {pp,rw,wr} x {Wl,Wr,bl}
    const float* Wlin = (const float*)d_in[8];
    const float* blin = (const float*)d_in[9];
    struct Sage { const float *Wl, *Wr, *bl; };
    Sage P[2][3];
    {
        int idx = 10;
        for (int l = 0; l < 2; ++l)
            for (int rel = 0; rel < 3; ++rel) {          // 0=pp, 1=rw, 2=wr
                P[l][rel].Wl = (const float*)d_in[idx++];
                P[l][rel].Wr = (const float*)d_in[idx++];
                P[l][rel].bl = (const float*)d_in[idx++];
            }
    }

    // workspace carve-up
    char* w = (char*)d_ws;
    float* msgA = (float*)w; w += (size_t)NP * DFEAT * 4;
    float* msgB = (float*)w; w += (size_t)NP * DFEAT * 4;
    float* cntA = (float*)w; w += (size_t)NP * 4;
    float* cntB = (float*)w; w += (size_t)NP * 4;
    float* hp0  = (float*)w; w += (size_t)NP * DFEAT * 4;
    float* hp1  = (float*)w; w += (size_t)NP * DFEAT * 4;
    float* ha0  = (float*)w; w += (size_t)NA * DFEAT * 4;
    float* ha1  = (float*)w; w += (size_t)NA * DFEAT * 4;
    float* Wrc[2]; float* blc[2];
    Wrc[0] = (float*)w; w += (size_t)DFEAT * DFEAT * 4;
    Wrc[1] = (float*)w; w += (size_t)DFEAT * DFEAT * 4;
    blc[0] = (float*)w; w += (size_t)DFEAT * 4;
    blc[1] = (float*)w; w += (size_t)DFEAT * 4;

    // Pre-sum the two root/self linears acting on x_paper (pp + wr relations).
    for (int l = 0; l < 2; ++l)
        combine_w_kernel<<<(DFEAT * DFEAT + 255) / 256, 256, 0, stream>>>(
            P[l][0].Wr, P[l][2].Wr, P[l][0].bl, P[l][2].bl, Wrc[l], blc[l]);

    for (int l = 0; l < 2; ++l) {
        const float* xp = (l == 0) ? x_paper  : hp0;
        const float* xa = (l == 0) ? x_author : ha0;
        float* yp = (l == 0) ? hp0 : hp1;
        float* ya = (l == 0) ? ha0 : ha1;

        // paper <- pp(paper) + wr(author)
        hipMemsetAsync(msgA, 0, (size_t)NP * DFEAT * 4, stream);
        hipMemsetAsync(msgB, 0, (size_t)NP * DFEAT * 4, stream);
        hipMemsetAsync(cntA, 0, (size_t)NP * 4, stream);
        hipMemsetAsync(cntB, 0, (size_t)NP * 4, stream);
        scatter_add_kernel<<<EPP / 8, 256, 0, stream>>>(xp, pp_s, pp_d, msgA, cntA, EPP);
        scatter_add_kernel<<<EWR / 8, 256, 0, stream>>>(xa, wr_s, wr_d, msgB, cntB, EWR);
        sage_gemm_kernel<<<NP / 16, 256, 0, stream>>>(
            msgA, cntA, P[l][0].Wl,
            msgB, cntB, P[l][2].Wl,
            xp, Wrc[l], blc[l], yp, NP);

        // author <- rw(paper)
        hipMemsetAsync(msgA, 0, (size_t)NA * DFEAT * 4, stream);
        hipMemsetAsync(cntA, 0, (size_t)NA * 4, stream);
        scatter_add_kernel<<<ERW / 8, 256, 0, stream>>>(xp, rw_s, rw_d, msgA, cntA, ERW);
        sage_gemm_kernel<<<NA / 16, 256, 0, stream>>>(
            msgA, cntA, P[l][1].Wl,
            nullptr, nullptr, nullptr,
            xa, P[l][1].Wr, P[l][1].bl, ya, NA);
    }

    final_linear_kernel<<<NP / 16, 32, 0, stream>>>(hp1, Wlin, blin, (float*)d_out, NP);
}